// TGCNLayer_68779606278978
// MI455X (gfx1250) — compile-verified
//
#include <hip/hip_runtime.h>
#include <math.h>

#define C 128          // IN_C == OUT_C == 128
#define TFEAT 16
#define EPSV 1e-5f
#define NEGSLOPE 0.01f
#define WSTRIDE 130    // padded row stride (floats) for transposed W in LDS

typedef __attribute__((ext_vector_type(2))) float v2f;
typedef __attribute__((ext_vector_type(8))) float v8f;

#if __has_builtin(__builtin_amdgcn_tensor_load_to_lds)
#define HAVE_TDM 1
#else
#define HAVE_TDM 0
#endif

// ---------------------------------------------------------------------------
// TDM: issue async load of a [rows x 128] f32 panel (row stride 128) into LDS
// offset `lds_byte_off`. Tail rows beyond `rows_remaining` zero-fill (OOB
// reads return 0 per ISA §8.2). Descriptor per ISA §8.3/8.4. Wave-uniform.
// ---------------------------------------------------------------------------
#if HAVE_TDM
__device__ __forceinline__ void tdm_issue_panel(const float* gsrc,
                                                unsigned lds_byte_off,
                                                int rows_remaining) {
  typedef __attribute__((ext_vector_type(4))) unsigned int tdm_u32x4;
  typedef __attribute__((ext_vector_type(8))) int tdm_i32x8;
  typedef __attribute__((ext_vector_type(4))) int tdm_i32x4;
  unsigned long long ga = (unsigned long long)(uintptr_t)gsrc;
  unsigned rows = (rows_remaining > 0) ? (unsigned)rows_remaining : 0u;

  tdm_u32x4 g0;
  g0.x = 1u;                                              // count=1 (valid user D#)
  g0.y = lds_byte_off;                                    // lds_addr
  g0.z = (unsigned)(ga & 0xFFFFFFFFu);                    // global_addr[31:0]
  g0.w = (unsigned)((ga >> 32) & 0x01FFFFFFu)             // global_addr[56:32]
       | (2u << 30);                                      // type = 2 ("image")

  tdm_i32x8 g1;
  g1[0] = (int)(2u << 16);                                // data_size = 2 (4 bytes)
  g1[1] = (int)(128u << 16);                              // tensor_dim0[15:0] = 128
  g1[2] = (int)((rows & 0xFFFFu) << 16);                  // dim0 hi | tensor_dim1 lo
  g1[3] = (int)(((rows >> 16) & 0xFFFFu) | (128u << 16)); // dim1 hi | tile_dim0 = 128
  g1[4] = 128;                                            // tile_dim1 = 128
  g1[5] = 128;                                            // tensor_dim0_stride = 128
  g1[6] = 0;
  g1[7] = 0;

  tdm_i32x4 z4 = {0, 0, 0, 0};                            // 2-D tensor: groups 2/3 unused
#if __clang_major__ >= 23
  tdm_i32x8 z8 = {0, 0, 0, 0, 0, 0, 0, 0};
  __builtin_amdgcn_tensor_load_to_lds(g0, g1, z4, z4, z8, 0);
#else
  __builtin_amdgcn_tensor_load_to_lds(g0, g1, z4, z4, 0);
#endif
}
#endif

// ---------------------------------------------------------------------------
// Prep: cvec[j] = sum_t sin(td * mf[t] * pi) * W_gcn[(128+t)*128 + j]
// (time encoding is identical for every node -> folds to a column bias)
// ---------------------------------------------------------------------------
__global__ void k_prep(const float* __restrict__ tdp, const float* __restrict__ mf,
                       const float* __restrict__ Wg, float* __restrict__ cvec) {
  int c = threadIdx.x;            // 128 threads
  float td = tdp[0];
  float acc = 0.f;
  #pragma unroll
  for (int t = 0; t < TFEAT; ++t) {
    float enc = sinf(td * mf[t] * 3.14159265358979323846f);
    acc += enc * Wg[(C + t) * C + c];
  }
  cvec[c] = acc;
}

// deg[i] = 1.0 (self loop weight), stats zeroed
__global__ void k_init(float* __restrict__ deg, float* __restrict__ stats, int n) {
  int i = blockIdx.x * blockDim.x + threadIdx.x;
  if (i < n) deg[i] = 1.0f;
  if (i < 2 * C) stats[i] = 0.f;
}

// deg[col[e]] += ew[e]
__global__ void k_deg(const int* __restrict__ ei, const float* __restrict__ ew,
                      float* __restrict__ deg, int ne) {
  int e = blockIdx.x * blockDim.x + threadIdx.x;
  if (e < ne) atomicAdd(&deg[ei[ne + e]], ew[e]);
}

// deg -> dinv in place
__global__ void k_dinv(float* __restrict__ deg, int n) {
  int i = blockIdx.x * blockDim.x + threadIdx.x;
  if (i >= n) return;
  float d = deg[i];
  deg[i] = (d > 0.f) ? rsqrtf(fmaxf(d, 1e-12f)) : 0.f;
}

// ---------------------------------------------------------------------------
// Shared fp32-WMMA GEMM core. Block = 128 threads (4 waves), 128 rows/block.
// Wave w owns rows [32w, 32w+32): two 16-row strips sharing every B frag.
// sA: [128][128] row-major at LDS offset 0 (TDM destination).
// sWt: W transposed, [128 cols][WSTRIDE] -> B frag = one aligned b64 read.
// ---------------------------------------------------------------------------
__device__ __forceinline__ void gemm_core(const float* __restrict__ sA,
                                          const float* __restrict__ sWt,
                                          int tid, v8f acc0[8], v8f acc1[8]) {
  int lane = tid & 31, wv = tid >> 5;
  int m  = lane & 15;
  int kh = (lane >> 4) << 1;                     // 0 or 2 (K half per ISA A layout)
  const float* aRow0 = sA + (wv * 32 + m) * C;
  const float* aRow1 = aRow0 + 16 * C;
  const float* bBase = sWt + (lane & 15) * WSTRIDE;
  for (int k0 = 0; k0 < C; k0 += 4) {
    v2f a0 = *(const v2f*)(aRow0 + k0 + kh);
    v2f a1 = *(const v2f*)(aRow1 + k0 + kh);
    const float* bp = bBase + k0 + kh;
    #pragma unroll
    for (int t = 0; t < 8; ++t) {
      v2f b = *(const v2f*)(bp + t * 16 * WSTRIDE);
      acc0[t] = __builtin_amdgcn_wmma_f32_16x16x4_f32(
          false, a0, false, b, (short)0, acc0[t], false, false);
      acc1[t] = __builtin_amdgcn_wmma_f32_16x16x4_f32(
          false, a1, false, b, (short)0, acc1[t], false, false);
    }
  }
}

__device__ __forceinline__ void gemm_store(float* __restrict__ Out,
                                           const float* __restrict__ addvec,
                                           int tid, int rowBase, int n,
                                           const v8f acc0[8], const v8f acc1[8]) {
  int lane = tid & 31, wv = tid >> 5;
  int ncol = lane & 15;
  int rloc = (lane >> 4) * 8;                    // C/D layout: VGPR i -> M = i + 8*(lane>=16)
  #pragma unroll
  for (int t = 0; t < 8; ++t) {
    int col = t * 16 + ncol;
    float add = addvec[col];
    #pragma unroll
    for (int i2 = 0; i2 < 8; ++i2) {
      int r0 = rowBase + wv * 32 + rloc + i2;
      if (r0 < n)      Out[(size_t)r0 * C + col] = acc0[t][i2] + add;
      int r1 = r0 + 16;
      if (r1 < n)      Out[(size_t)r1 * C + col] = acc1[t][i2] + add;
    }
  }
}

// GEMM 1: h0 = x @ W_gcn[0:128,:] + cvec
__global__ void __launch_bounds__(128)
k_gemm_gcn(const float* __restrict__ A, const float* __restrict__ W,
           const float* __restrict__ cvec, float* __restrict__ Out, int n) {
  extern __shared__ float smem[];
  float* sA  = smem;                 // 128*128 floats @ LDS offset 0
  float* sWt = smem + C * C;         // 128*WSTRIDE floats
  int tid = threadIdx.x;
  int rowBase = blockIdx.x * 128;

#if HAVE_TDM
  if (tid < 32) tdm_issue_panel(A + (size_t)rowBase * C, 0u, n - rowBase);
#endif
  // stage W transposed while the TDM is in flight
  for (int i = tid; i < C * C; i += 128) {
    sWt[(i & 127) * WSTRIDE + (i >> 7)] = W[i];
  }
#if HAVE_TDM
  if (tid < 32) __builtin_amdgcn_s_wait_tensorcnt(0);
#else
  for (int i = tid; i < 128 * C; i += 128) {
    int r = rowBase + (i >> 7);
    sA[i] = (r < n) ? A[(size_t)r * C + (i & 127)] : 0.f;
  }
#endif
  __syncthreads();

  v8f acc0[8] = {}, acc1[8] = {};
  gemm_core(sA, sWt, tid, acc0, acc1);
  gemm_store(Out, cvec, tid, rowBase, n, acc0, acc1);
}

// h1[i][c] = dinv[i]^2 * h0[i][c] + b_gcn[c]   (self-loop norm = dinv[i]*1*dinv[i])
__global__ void k_selfloop(const float* __restrict__ h0, const float* __restrict__ dinv,
                           const float* __restrict__ bg, float* __restrict__ h1, int n) {
  size_t gid = (size_t)blockIdx.x * blockDim.x + threadIdx.x;
  if (gid >= (size_t)n * C) return;
  int i = (int)(gid >> 7);
  int c = (int)(gid & 127);
  float dv = dinv[i];
  h1[gid] = dv * dv * h0[gid] + bg[c];
}

// One wave32 per edge: h1[col] += dinv[row]*w*dinv[col] * h0[row]
// b128 gathers: lane loads float4, issues 4 f32 atomics.
__global__ void k_scatter(const int* __restrict__ ei, const float* __restrict__ ew,
                          const float* __restrict__ dinv, const float* __restrict__ h0,
                          float* __restrict__ h1, int ne) {
  int gid = blockIdx.x * blockDim.x + threadIdx.x;
  int e = gid >> 5, lane = gid & 31;
  if (e >= ne) return;
  int row = ei[e];
  int col = ei[ne + e];
  float norm = dinv[row] * ew[e] * dinv[col];
  const float4* src4 = (const float4*)(h0 + (size_t)row * C);
  float*        dst  = h1 + (size_t)col * C + lane * 4;
  float4 v = src4[lane];
  atomicAdd(dst + 0, norm * v.x);
  atomicAdd(dst + 1, norm * v.y);
  atomicAdd(dst + 2, norm * v.z);
  atomicAdd(dst + 3, norm * v.w);
}

// Per-channel sum / sumsq over N rows (thread = channel, block = row chunk)
__global__ void k_stats(const float* __restrict__ h1, float* __restrict__ stats, int n) {
  int c = threadIdx.x;            // 128 threads
  int per = (n + gridDim.x - 1) / (int)gridDim.x;
  int r0 = blockIdx.x * per;
  int r1 = r0 + per; if (r1 > n) r1 = n;
  float s = 0.f, s2 = 0.f;
  for (int r = r0; r < r1; ++r) {
    float v = h1[(size_t)r * C + c];
    s += v; s2 += v * v;
  }
  atomicAdd(&stats[c], s);
  atomicAdd(&stats[C + c], s2);
}

// bnA[c] = gamma*rsqrt(var+eps); bnD[c] = beta - mean*bnA
__global__ void k_bnparam(const float* __restrict__ stats, const float* __restrict__ gamma,
                          const float* __restrict__ beta, float* __restrict__ bnA,
                          float* __restrict__ bnD, int n) {
  int c = threadIdx.x;            // 128 threads
  float inv_n = 1.0f / (float)n;
  float mean = stats[c] * inv_n;
  float var  = stats[C + c] * inv_n - mean * mean;
  float a = gamma[c] * rsqrtf(var + EPSV);
  bnA[c] = a;
  bnD[c] = beta[c] - mean * a;
}

// ---------------------------------------------------------------------------
// GEMM 2 (in-place on d_out): out = LeakyReLU(bnA*h + bnD) @ W_lin + b_lin
// TDM stages the raw panel; BN affine + LeakyReLU applied in-place in LDS.
// Each block reads only its own rows (TDM zero-fills the tail) and all its
// global reads precede its writes -> in-place is race-free.
// ---------------------------------------------------------------------------
__global__ void __launch_bounds__(128)
k_gemm_out(float* __restrict__ H, const float* __restrict__ W,
           const float* __restrict__ bnA, const float* __restrict__ bnD,
           const float* __restrict__ blin, int n) {
  extern __shared__ float smem[];
  float* sA  = smem;
  float* sWt = smem + C * C;
  int tid = threadIdx.x;
  int rowBase = blockIdx.x * 128;

#if HAVE_TDM
  if (tid < 32) tdm_issue_panel(H + (size_t)rowBase * C, 0u, n - rowBase);
#endif
  for (int i = tid; i < C * C; i += 128) {
    sWt[(i & 127) * WSTRIDE + (i >> 7)] = W[i];
  }
#if HAVE_TDM
  if (tid < 32) __builtin_amdgcn_s_wait_tensorcnt(0);
#else
  for (int i = tid; i < 128 * C; i += 128) {
    int r = rowBase + (i >> 7);
    sA[i] = (r < n) ? H[(size_t)r * C + (i & 127)] : 0.f;
  }
#endif
  __syncthreads();

  // BN affine + LeakyReLU in place on the staged panel
  for (int i = tid; i < 128 * C; i += 128) {
    int cc = i & 127;
    float v = bnA[cc] * sA[i] + bnD[cc];
    sA[i] = (v >= 0.f) ? v : NEGSLOPE * v;
  }
  __syncthreads();

  v8f acc0[8] = {}, acc1[8] = {};
  gemm_core(sA, sWt, tid, acc0, acc1);
  gemm_store(H, blin, tid, rowBase, n, acc0, acc1);
}

// ---------------------------------------------------------------------------
extern "C" void kernel_launch(void* const* d_in, const int* in_sizes, int n_in,
                              void* d_out, int out_size, void* d_ws, size_t ws_size,
                              hipStream_t stream) {
  const float* x     = (const float*)d_in[0];
  const int*   ei    = (const int*)  d_in[1];
  const float* ew    = (const float*)d_in[2];
  const float* tdp   = (const float*)d_in[3];
  const float* mf    = (const float*)d_in[4];
  const float* Wg    = (const float*)d_in[5];
  const float* bg    = (const float*)d_in[6];
  const float* gamma = (const float*)d_in[7];
  const float* beta  = (const float*)d_in[8];
  const float* Wl    = (const float*)d_in[9];
  const float* bl    = (const float*)d_in[10];
  float* out = (float*)d_out;

  int n = in_sizes[0] / C;      // 50000
  int e = in_sizes[2];          // 500000

  float* ws   = (float*)d_ws;
  float* h0   = ws;                         // n*C
  float* deg  = ws + (size_t)n * C;         // n (becomes dinv in place)
  float* cvec = deg + n;                    // C
  float* stat = cvec + C;                   // 2*C (sum, sumsq)
  float* bnA  = stat + 2 * C;               // C
  float* bnD  = bnA + C;                    // C

  size_t shmem = (size_t)(C * C + C * WSTRIDE) * sizeof(float);  // ~129 KB
  size_t tot = (size_t)n * C;

  k_prep    <<<1, C, 0, stream>>>(tdp, mf, Wg, cvec);
  k_init    <<<(n + 255) / 256, 256, 0, stream>>>(deg, stat, n);
  k_deg     <<<(e + 255) / 256, 256, 0, stream>>>(ei, ew, deg, e);
  k_dinv    <<<(n + 255) / 256, 256, 0, stream>>>(deg, n);
  k_gemm_gcn<<<(n + 127) / 128, 128, shmem, stream>>>(x, Wg, cvec, h0, n);
  k_selfloop<<<(int)((tot + 255) / 256), 256, 0, stream>>>(h0, deg, bg, out, n);
  k_scatter <<<(int)(((size_t)e * 32 + 255) / 256), 256, 0, stream>>>(ei, ew, deg, h0, out, e);
  k_stats   <<<512, C, 0, stream>>>(out, stat, n);
  k_bnparam <<<1, C, 0, stream>>>(stat, gamma, beta, bnA, bnD, n);
  k_gemm_out<<<(n + 127) / 128, 128, shmem, stream>>>(out, Wl, bnA, bnD, bl, n);
}